// MultiHeadLinearAttention_34445637714089
// MI455X (gfx1250) — compile-verified
//
#include <hip/hip_runtime.h>
#include <hip/hip_bf16.h>

typedef __attribute__((ext_vector_type(2))) float v2f;
typedef __attribute__((ext_vector_type(4))) float v4f;
typedef __attribute__((ext_vector_type(8))) float v8f;
typedef __attribute__((ext_vector_type(4))) unsigned int v4u;
typedef __attribute__((ext_vector_type(8))) int v8i;
typedef __attribute__((ext_vector_type(4))) int v4i;

#define WMMA_F32(a, b, c) \
    __builtin_amdgcn_wmma_f32_16x16x4_f32(false, (a), false, (b), (short)0, (c), false, false)

// Problem constants
constexpr int B_   = 4;
constexpr int T_   = 4096;
constexpr int S_   = 4096;
constexpr int D_   = 512;     // D_MODEL = D_KEY = D_VALUE
constexpr int H_   = 8;
constexpr int DH_  = 64;      // per-head dim
constexpr int NOM_ = 64;      // N_OMEGA
constexpr int F_   = 128;     // 2*N_OMEGA feature dim
constexpr int DV1_ = 65;      // DV_H + 1 (ones column)
constexpr int BT_  = B_ * T_; // 16384 rows
constexpr float EPS_ = 1e-9f;
constexpr float RS_  = 0.08838834764831845f; // 1/sqrt(2*N_OMEGA)
constexpr int KV_SPLIT = 16;

static __device__ __forceinline__ v8f zero8() {
    v8f z = {0.f, 0.f, 0.f, 0.f, 0.f, 0.f, 0.f, 0.f};
    return z;
}

// --- CDNA5 async/TDM helpers ------------------------------------------------
// Low 32 bits of a generic pointer to LDS == LDS byte offset (LLVM builds the
// generic pointer as {aperture_hi, offset32}).
static __device__ __forceinline__ unsigned int lds_addr32(const void* p) {
    return (unsigned int)(unsigned long long)(uintptr_t)p;
}

// Async memory -> LDS copy, 16 bytes per lane. Tracked by ASYNCcnt.
static __device__ __forceinline__ void async_b128(const float* gsrc, float* ldst) {
    asm volatile("global_load_async_to_lds_b128 %0, %1, off"
                 :
                 : "v"(lds_addr32(ldst)), "v"(gsrc)
                 : "memory");
}

template <int N>
static __device__ __forceinline__ void wait_asynccnt() {
    asm volatile("s_wait_asynccnt %0" : : "i"(N) : "memory");
}

// Tensor Data Mover: 2D tile (rows x cols of f32, row stride in elems) -> LDS.
// D# packing per CDNA5 ISA 8.3/8.4: group0 = {count=1, lds_addr, global_addr,
// type=2}; group1 = {mask=0, data_size=4B, tensor_dim0/1, tile_dim0/1, stride0}.
// This toolchain exposes the 6-arg builtin: (g0, g1, g2, g3, g_extra, cpol).
static __device__ __forceinline__ void tdm_load_2d(const float* gsrc, float* ldst,
                                                   unsigned rows, unsigned cols,
                                                   unsigned row_stride) {
    const unsigned long long ga = (unsigned long long)(uintptr_t)gsrc;
    v4u g0;
    g0.x = 1u;                                   // count=1 (valid user D#)
    g0.y = lds_addr32(ldst);                     // lds_addr [63:32]
    g0.z = (unsigned int)(ga & 0xFFFFFFFFu);     // global_addr[31:0]
    g0.w = (unsigned int)((ga >> 32) & 0x01FFFFFFu) | (2u << 30); // addr[56:32], type=2
    v8i g1;
    g1[0] = (int)(2u << 16);                                          // mask=0, data_size=4B
    g1[1] = (int)((cols & 0xFFFFu) << 16);                            // tensor_dim0 lo16
    g1[2] = (int)(((cols >> 16) & 0xFFFFu) | ((rows & 0xFFFFu) << 16)); // dim0 hi | dim1 lo
    g1[3] = (int)(((rows >> 16) & 0xFFFFu) | ((cols & 0xFFFFu) << 16)); // dim1 hi | tile_dim0
    g1[4] = (int)(rows & 0xFFFFu);                                    // tile_dim1 | tile_dim2=0
    g1[5] = (int)row_stride;                                          // tensor_dim0_stride lo32
    g1[6] = 0;                                                        // stride0 hi | stride1 lo
    g1[7] = 0;
    const v4i z4 = {0, 0, 0, 0};
    const v8i z8 = {0, 0, 0, 0, 0, 0, 0, 0};
    __builtin_amdgcn_tensor_load_to_lds(g0, g1, z4, z4, z8, 0);
}

// ---------------------------------------------------------------------------
// Kernel 1: per-head projection GEMM.  Y[row, h*64+d] = sum_f X[row,f]*Wh[h,f,d]
// 64x64 block tile, K staged 16/chunk via double-buffered async-to-LDS.
// 8 waves: 4 row-groups x 2 col-groups, 2 C-tiles each.
// ---------------------------------------------------------------------------
__global__ __launch_bounds__(256) void gemm_proj(const float* __restrict__ X,
                                                 const float* __restrict__ Wh,
                                                 float* __restrict__ Y) {
    __shared__ __align__(16) float Xs[2][64][20];   // pad 20: rows 16B-aligned, banks 20r+c distinct
    __shared__ __align__(16) float Ws[2][16][64];

    const int tid  = threadIdx.x;
    const int lane = tid & 31;
    const int wid  = tid >> 5;
    const int wr   = wid >> 1;        // row group 0..3
    const int wc   = wid & 1;         // col group 0..1
    const int r    = lane & 15;
    const int kh   = (lane >> 4) * 2; // K sub-offset 0 or 2

    const int row0 = blockIdx.x * 64;
    const int h    = blockIdx.y;      // col tile == head

    const int lr   = tid >> 2;        // 0..63 (Xs row)
    const int lc   = (tid & 3) * 4;   // 0,4,8,12
    const int kr   = tid >> 4;        // 0..15 (Ws row)
    const int wcol = (tid & 15) * 4;  // 0..60

    auto issue = [&](int buf, int k0) {
        async_b128(X + (size_t)(row0 + lr) * D_ + k0 + lc, &Xs[buf][lr][lc]);
        async_b128(Wh + ((size_t)(h * D_ + k0 + kr)) * DH_ + wcol, &Ws[buf][kr][wcol]);
    };

    v8f acc[2] = {zero8(), zero8()};
    issue(0, 0);

    for (int kc = 0; kc < D_ / 16; ++kc) {
        const int cur = kc & 1;
        if (kc < D_ / 16 - 1) {
            issue(cur ^ 1, (kc + 1) * 16);
            wait_asynccnt<2>();   // this chunk's 2 loads done; next chunk in flight
        } else {
            wait_asynccnt<0>();
        }
        __syncthreads();

#pragma unroll
        for (int ks = 0; ks < 4; ++ks) {
            const int kk = ks * 4;
            v2f a;
            a.x = Xs[cur][wr * 16 + r][kk + kh + 0];
            a.y = Xs[cur][wr * 16 + r][kk + kh + 1];
#pragma unroll
            for (int j = 0; j < 2; ++j) {
                const int c = wc * 32 + j * 16 + r;
                v2f b;
                b.x = Ws[cur][kk + kh + 0][c];
                b.y = Ws[cur][kk + kh + 1][c];
                acc[j] = WMMA_F32(a, b, acc[j]);
            }
        }
        __syncthreads();
    }

#pragma unroll
    for (int j = 0; j < 2; ++j) {
#pragma unroll
        for (int i = 0; i < 8; ++i) {
            const int m   = (lane < 16) ? i : (8 + i);
            const int row = row0 + wr * 16 + m;
            const int col = h * 64 + wc * 32 + j * 16 + (lane & 15);
            Y[(size_t)row * D_ + col] = acc[j][i];
        }
    }
}

// ---------------------------------------------------------------------------
// Kernel 2: random-feature map phi.
// xw = x_tile(16x64) @ omega^T(64x64) via WMMA; ss = 0.5*||x||^2 per row;
// out = (exp(+-xw - ss) + eps)/sqrt(128).  8 waves = 8 heads of one row-tile.
// x tiles staged via async-to-LDS; omega transposed manually (one-time, tiny).
// ---------------------------------------------------------------------------
__global__ __launch_bounds__(256) void phi_kernel(const float* __restrict__ Xall,
                                                  const float* __restrict__ omega,
                                                  float* __restrict__ Phi) {
    __shared__ __align__(16) float omT[64][65];      // omT[k][m] = omega[m][k] (scalar stores)
    __shared__ __align__(16) float xs[8][16][68];    // pad 68: 16B rows, banks 4r+c distinct
    __shared__ float ss[8][16];

    const int tid  = threadIdx.x;
    const int lane = tid & 31;
    const int h    = tid >> 5;           // wave id == head
    const int r    = lane & 15;
    const int half = lane >> 4;
    const int kh   = half * 2;

    const int gr0 = blockIdx.x * 16;     // first global row (over B*T)

    // Async-stage per-head x tile: lane loads 32 floats of row r (8 x b128).
#pragma unroll
    for (int j4 = 0; j4 < 8; ++j4) {
        const int c = half * 32 + j4 * 4;
        async_b128(Xall + (size_t)(gr0 + r) * D_ + h * 64 + c, &xs[h][r][c]);
    }
    // Stage omega transposed (4K floats, one-time).
    {
        const int m  = tid >> 2;
        const int k0 = (tid & 3) * 16;
#pragma unroll
        for (int j = 0; j < 4; ++j) {
            const v4f ov = *(const v4f*)(omega + (size_t)m * DH_ + k0 + j * 4);
            omT[k0 + j * 4 + 0][m] = ov.x;
            omT[k0 + j * 4 + 1][m] = ov.y;
            omT[k0 + j * 4 + 2][m] = ov.z;
            omT[k0 + j * 4 + 3][m] = ov.w;
        }
    }
    wait_asynccnt<0>();
    __syncthreads();

    if (lane < 16) {
        float s = 0.f;
#pragma unroll
        for (int k = 0; k < 64; ++k) {
            const float x = xs[h][lane][k];
            s += x * x;
        }
        ss[h][lane] = 0.5f * s;
    }
    __syncthreads();

    v8f acc[4] = {zero8(), zero8(), zero8(), zero8()};
#pragma unroll
    for (int ks = 0; ks < 16; ++ks) {
        const int kk = ks * 4;
        v2f a;
        a.x = xs[h][r][kk + kh + 0];
        a.y = xs[h][r][kk + kh + 1];
#pragma unroll
        for (int jt = 0; jt < 4; ++jt) {
            const int c = jt * 16 + r;
            v2f b;
            b.x = omT[kk + kh + 0][c];
            b.y = omT[kk + kh + 1][c];
            acc[jt] = WMMA_F32(a, b, acc[jt]);
        }
    }

#pragma unroll
    for (int jt = 0; jt < 4; ++jt) {
        const int m = jt * 16 + (lane & 15);
#pragma unroll
        for (int i = 0; i < 8; ++i) {
            const int mr  = (lane < 16) ? i : (8 + i);
            const int gr  = gr0 + mr;
            const int b   = gr >> 12;        // / T_
            const int t   = gr & (T_ - 1);
            const float s = ss[h][mr];
            const float xw = acc[jt][i];
            const size_t base = ((size_t)(b * H_ + h) * T_ + t) * F_;
            Phi[base + m]        = (__expf(xw - s)  + EPS_) * RS_;
            Phi[base + NOM_ + m] = (__expf(-xw - s) + EPS_) * RS_;
        }
    }
}

// ---------------------------------------------------------------------------
// Kernel 3: kv partials.  kv[bh][m][d] = sum_s kp[bh,s,m] * v1[bh,s,d]
// Grid (B*H, KV_SPLIT), 32-deep S chunks, double buffered:
//   kp tile (32x128 f32) via TDM tensor_load_to_lds (wave 0, TENSORcnt),
//   v tile  (cols 0..63) via async-to-LDS (all waves, ASYNCcnt),
//   cols 64..79 (ones + pad) written once per buffer.
// 8 waves = 8 m-tiles, 5 N-tile accumulators each.
// ---------------------------------------------------------------------------
__global__ __launch_bounds__(256) void kv_kernel(const float* __restrict__ Kp,
                                                 const float* __restrict__ Vall,
                                                 float* __restrict__ Part) {
    __shared__ __align__(16) float kps[2][32][128];  // [s][m]
    __shared__ __align__(16) float vs[2][32][80];    // [s][d], col 64 = 1, 65..79 = 0

    const int tid  = threadIdx.x;
    const int lane = tid & 31;
    const int wid  = tid >> 5;       // m-tile 0..7
    const int r    = lane & 15;
    const int kh   = (lane >> 4) * 2;

    const int bh = blockIdx.x;       // 0..31
    const int b  = bh / H_;
    const int h  = bh % H_;
    const int s0 = blockIdx.y * (S_ / KV_SPLIT);

    // Constant columns of both v buffers (never touched by async engine).
    for (int idx = tid; idx < 2 * 32 * 16; idx += 256) {
        const int bufi = idx >> 9;             // /512
        const int rem  = idx & 511;
        const int sr   = rem >> 4;
        const int c    = 64 + (rem & 15);
        vs[bufi][sr][c] = (c == 64) ? 1.0f : 0.0f;
    }

    const int sr8 = tid >> 3;        // 0..31
    const int c8  = (tid & 7) * 8;   // 0..56

    auto issue = [&](int buf, int sb) {
        if (wid == 0) {
            tdm_load_2d(Kp + ((size_t)bh * S_ + sb) * F_, &kps[buf][0][0],
                        /*rows=*/32, /*cols=*/128, /*row_stride=*/128);
        }
        const float* src = Vall + ((size_t)(b * S_) + sb + sr8) * D_ + h * 64 + c8;
        async_b128(src,     &vs[buf][sr8][c8]);
        async_b128(src + 4, &vs[buf][sr8][c8 + 4]);
    };

    v8f acc[5] = {zero8(), zero8(), zero8(), zero8(), zero8()};
    issue(0, s0);

    constexpr int NCHUNK = (S_ / KV_SPLIT) / 32;   // 8
    for (int sc = 0; sc < NCHUNK; ++sc) {
        const int cur = sc & 1;
        if (sc < NCHUNK - 1) {
            issue(cur ^ 1, s0 + (sc + 1) * 32);
            if (wid == 0) __builtin_amdgcn_s_wait_tensorcnt(1);
            wait_asynccnt<2>();
        } else {
            if (wid == 0) __builtin_amdgcn_s_wait_tensorcnt(0);
            wait_asynccnt<0>();
        }
        __syncthreads();

#pragma unroll
        for (int ks = 0; ks < 8; ++ks) {
            const int kk = ks * 4;
            v2f a;  // A[m][s] = kp[s][m]
            a.x = kps[cur][kk + kh + 0][wid * 16 + r];
            a.y = kps[cur][kk + kh + 1][wid * 16 + r];
#pragma unroll
            for (int jt = 0; jt < 5; ++jt) {
                v2f bfr;
                bfr.x = vs[cur][kk + kh + 0][jt * 16 + r];
                bfr.y = vs[cur][kk + kh + 1][jt * 16 + r];
                acc[jt] = WMMA_F32(a, bfr, acc[jt]);
            }
        }
        __syncthreads();
    }

    const size_t pbase = ((size_t)bh * KV_SPLIT + blockIdx.y) * (F_ * DV1_);
#pragma unroll
    for (int jt = 0; jt < 5; ++jt) {
        const int d = jt * 16 + (lane & 15);
        if (d < DV1_) {
#pragma unroll
            for (int i = 0; i < 8; ++i) {
                const int m = wid * 16 + ((lane < 16) ? i : (8 + i));
                Part[pbase + (size_t)m * DV1_ + d] = acc[jt][i];
            }
        }
    }
}

// Deterministic split-S reduction.
__global__ __launch_bounds__(256) void kv_reduce(const float* __restrict__ Part,
                                                 float* __restrict__ Kv) {
    const int idx = blockIdx.x * 256 + threadIdx.x;
    const int total = B_ * H_ * F_ * DV1_;
    if (idx >= total) return;
    const int bh  = idx / (F_ * DV1_);
    const int rem = idx % (F_ * DV1_);
    float s = 0.f;
#pragma unroll
    for (int sp = 0; sp < KV_SPLIT; ++sp) {
        s += Part[((size_t)bh * KV_SPLIT + sp) * (F_ * DV1_) + rem];
    }
    Kv[idx] = s;
}

// ---------------------------------------------------------------------------
// Kernel 4: qkv = qp(16x128) @ kv(128x65); divide by normalizer column.
// qp tile async-staged to LDS; kv read from global (L2/WGP$-hot, reused 256x).
// Normalizer broadcast via __shfl from lanes 0/16 of N-tile 4.
// Output uses the reference's flat reshape: idx = ((b*H+h)*T + t)*64 + d.
// ---------------------------------------------------------------------------
__global__ __launch_bounds__(128) void out_kernel(const float* __restrict__ Qp,
                                                  const float* __restrict__ Kv,
                                                  float* __restrict__ Out) {
    __shared__ __align__(16) float qs[4][16][132];  // pad 132: 16B rows, banks 4r+c distinct

    const int tid  = threadIdx.x;
    const int lane = tid & 31;
    const int wid  = tid >> 5;       // 0..3
    const int r    = lane & 15;
    const int half = lane >> 4;
    const int kh   = half * 2;

    const int task = blockIdx.x * 4 + wid;   // 0..8191
    const int tt   = task % (T_ / 16);
    const int bh   = task / (T_ / 16);       // 0..31
    const int t0   = tt * 16;

    // Async-stage qp tile: lane loads 64 floats of row r (16 x b128).
    {
        const float* src = Qp + ((size_t)bh * T_ + t0 + r) * F_ + half * 64;
#pragma unroll
        for (int j = 0; j < 16; ++j) {
            async_b128(src + j * 4, &qs[wid][r][half * 64 + j * 4]);
        }
    }
    wait_asynccnt<0>();
    __syncthreads();

    v8f acc[5] = {zero8(), zero8(), zero8(), zero8(), zero8()};
    const float* kvb = Kv + (size_t)bh * F_ * DV1_;

#pragma unroll 4
    for (int ks = 0; ks < 32; ++ks) {
        const int kk = ks * 4;
        v2f a;
        a.x = qs[wid][r][kk + kh + 0];
        a.y = qs[wid][r][kk + kh + 1];
#pragma unroll
        for (int jt = 0; jt < 5; ++jt) {
            const int c = jt * 16 + r;
            v2f bfr;
            bfr.x = (c < DV1_) ? kvb[(size_t)(kk + kh + 0) * DV1_ + c] : 0.0f;
            bfr.y = (c < DV1_) ? kvb[(size_t)(kk + kh + 1) * DV1_ + c] : 0.0f;
            acc[jt] = WMMA_F32(a, bfr, acc[jt]);
        }
    }

#pragma unroll
    for (int i = 0; i < 8; ++i) {
        const float norm = __shfl(acc[4][i], lane & 16);
        const float inv  = 1.0f / norm;
        const int t = t0 + ((lane < 16) ? i : (8 + i));
        const size_t base = ((size_t)bh * T_ + t) * 64;
#pragma unroll
        for (int jt = 0; jt < 4; ++jt) {
            Out[base + jt * 16 + (lane & 15)] = acc[jt][i] * inv;
        }
    }
}

// ---------------------------------------------------------------------------
extern "C" void kernel_launch(void* const* d_in, const int* in_sizes, int n_in,
                              void* d_out, int out_size, void* d_ws, size_t ws_size,
                              hipStream_t stream) {
    const float* query = (const float*)d_in[0];
    const float* value = (const float*)d_in[1];
    const float* key   = (const float*)d_in[2];
    const float* wq    = (const float*)d_in[3];
    const float* wv    = (const float*)d_in[4];
    const float* wk    = (const float*)d_in[5];
    const float* omega = (const float*)d_in[6];
    float* out = (float*)d_out;

    float* ws    = (float*)d_ws;
    float* q_all = ws;                                   // 16384*512
    float* k_all = q_all + (size_t)BT_ * D_;
    float* v_all = k_all + (size_t)BT_ * D_;
    float* qp    = v_all + (size_t)BT_ * D_;             // 4*8*4096*128
    float* kp    = qp + (size_t)B_ * H_ * T_ * F_;
    float* part  = kp + (size_t)B_ * H_ * S_ * F_;       // 32*16*128*65
    float* kv    = part + (size_t)B_ * H_ * KV_SPLIT * F_ * DV1_;

    dim3 gemm_grid(BT_ / 64, D_ / 64);  // 256 x 8
    gemm_proj<<<gemm_grid, 256, 0, stream>>>(query, wq, q_all);
    gemm_proj<<<gemm_grid, 256, 0, stream>>>(key,   wk, k_all);
    gemm_proj<<<gemm_grid, 256, 0, stream>>>(value, wv, v_all);

    phi_kernel<<<BT_ / 16, 256, 0, stream>>>(q_all, omega, qp);
    phi_kernel<<<BT_ / 16, 256, 0, stream>>>(k_all, omega, kp);

    kv_kernel<<<dim3(B_ * H_, KV_SPLIT), 256, 0, stream>>>(kp, v_all, part);

    const int kv_total = B_ * H_ * F_ * DV1_;            // 266240
    kv_reduce<<<(kv_total + 255) / 256, 256, 0, stream>>>(part, kv);

    out_kernel<<<(B_ * H_ * (T_ / 16)) / 4, 128, 0, stream>>>(qp, kv, out);
}